// GumbelTopK_22969485099581
// MI455X (gfx1250) — compile-verified
//
#include <hip/hip_runtime.h>
#include <hip/hip_bf16.h>
#include <stdint.h>

// Problem constants (from reference)
#define KSEL     30
#define MDIM     2048
#define ROWS_ALL (64 * 512)
#define RPB      8            // rows per block
#define TPB      256          // threads per block (8 waves of 32)
#define NWAVE    (TPB / 32)
#define EPSV     1e-20f
#define INV_SOFT 100.0f       // 1 / SOFTNESS

// Order-preserving bijection float -> uint (total order matches float order)
__device__ __forceinline__ unsigned orderKey(float f) {
    unsigned u = __float_as_uint(f);
    return (u & 0x80000000u) ? ~u : (u | 0x80000000u);
}
__device__ __forceinline__ float keyToFloat(unsigned k) {
    unsigned u = (k & 0x80000000u) ? (k & 0x7FFFFFFFu) : ~k;
    return __uint_as_float(u);
}

__global__ __launch_bounds__(TPB)
void GumbelTopK_kernel(const float* __restrict__ logits,
                       const float* __restrict__ U,
                       float* __restrict__ out)
{
    __shared__ float ldsL[2][MDIM];   // logits row, double buffered (16 KB)
    __shared__ float ldsU[2][MDIM];   // U row, double buffered      (16 KB)
    __shared__ float    redF[NWAVE];
    __shared__ unsigned redU[NWAVE];

    const int tid  = threadIdx.x;
    const int lane = tid & 31;
    const int wave = tid >> 5;
    const int rowBase = blockIdx.x * RPB;

    // Async prefetch of one row (logits+U) into LDS buffer `buf`.
    // 2048 floats = 8192 B = 512 x 16B chunks per array; 256 lanes x 2 chunks.
    auto prefetch = [&](int row, int buf) {
        const float* lrow = logits + (size_t)row * MDIM;
        const float* urow = U      + (size_t)row * MDIM;
        unsigned off0 = (unsigned)tid * 16u;       // bytes
        unsigned off1 = off0 + (unsigned)(TPB * 16); // +4096 B
        unsigned dL = (unsigned)(size_t)&ldsL[buf][0];
        unsigned dU = (unsigned)(size_t)&ldsU[buf][0];
        asm volatile("global_load_async_to_lds_b128 %0, %1, %2 offset:0"
                     :: "v"(dL + off0), "v"(off0), "s"(lrow) : "memory");
        asm volatile("global_load_async_to_lds_b128 %0, %1, %2 offset:0"
                     :: "v"(dL + off1), "v"(off1), "s"(lrow) : "memory");
        asm volatile("global_load_async_to_lds_b128 %0, %1, %2 offset:0"
                     :: "v"(dU + off0), "v"(off0), "s"(urow) : "memory");
        asm volatile("global_load_async_to_lds_b128 %0, %1, %2 offset:0"
                     :: "v"(dU + off1), "v"(off1), "s"(urow) : "memory");
    };

    prefetch(rowBase, 0);

    for (int r = 0; r < RPB; ++r) {
        const int row = rowBase + r;
        const int buf = r & 1;

        // Wait for this row's async loads, then barrier so every wave sees LDS.
        asm volatile("s_wait_asynccnt 0x0" ::: "memory");
        __syncthreads();

        // Kick off the next row's DMA into the other buffer (overlaps compute).
        if (r + 1 < RPB) prefetch(row + 1, buf ^ 1);

        // ---- load my 8 elements from LDS ----
        const float4* Lv = (const float4*)&ldsL[buf][tid * 8];
        const float4* Uv = (const float4*)&ldsU[buf][tid * 8];
        float4 l0 = Lv[0], l1 = Lv[1];
        float4 u0 = Uv[0], u1 = Uv[1];
        float lv[8] = {l0.x, l0.y, l0.z, l0.w, l1.x, l1.y, l1.z, l1.w};
        float uv[8] = {u0.x, u0.y, u0.z, u0.w, u1.x, u1.y, u1.z, u1.w};

        // ---- Gumbel perturbation (TAU == 1) + local max ----
        float p[8];
        float mx = -INFINITY;
#pragma unroll
        for (int j = 0; j < 8; ++j) {
            float g = -__logf(-__logf(uv[j] + EPSV) + EPSV);
            p[j] = lv[j] + g;
            mx = fmaxf(mx, p[j]);
        }

        // ---- block-wide max ----
#pragma unroll
        for (int o = 16; o; o >>= 1) mx = fmaxf(mx, __shfl_xor(mx, o, 32));
        if (lane == 0) redF[wave] = mx;
        __syncthreads();
        float rowmax = redF[0];
#pragma unroll
        for (int w = 1; w < NWAVE; ++w) rowmax = fmaxf(rowmax, redF[w]);
        __syncthreads();

        // ---- exp + block-wide sum ----
        float e[8];
        float s = 0.0f;
#pragma unroll
        for (int j = 0; j < 8; ++j) { e[j] = __expf(p[j] - rowmax); s += e[j]; }
#pragma unroll
        for (int o = 16; o; o >>= 1) s += __shfl_xor(s, o, 32);
        if (lane == 0) redF[wave] = s;
        __syncthreads();
        float rowsum = 0.0f;
#pragma unroll
        for (int w = 0; w < NWAVE; ++w) rowsum += redF[w];
        __syncthreads();
        float rinv = __frcp_rn(rowsum);

        // ---- K-th largest of perturbed values via bitwise binary search ----
        unsigned kk[8];
#pragma unroll
        for (int j = 0; j < 8; ++j) kk[j] = orderKey(p[j]);

        unsigned lo = 0u, hi = 0xFFFFFFFFu;
        for (int it = 0; it < 32 && lo < hi; ++it) {
            unsigned d   = hi - lo;
            unsigned mid = lo + (d >> 1) + (d & 1u);   // ceil midpoint, no overflow
            int c = 0;
#pragma unroll
            for (int j = 0; j < 8; ++j) c += (kk[j] >= mid) ? 1 : 0;
#pragma unroll
            for (int o = 16; o; o >>= 1) c += __shfl_xor(c, o, 32);
            if (lane == 0) redU[wave] = (unsigned)c;
            __syncthreads();
            unsigned tot = 0;
#pragma unroll
            for (int w = 0; w < NWAVE; ++w) tot += redU[w];
            __syncthreads();
            if (tot >= (unsigned)KSEL) lo = mid; else hi = mid - 1u;
        }
        // threshold prob = softmax(kth perturbed)  (monotone map preserves k-th)
        float thr = __expf(keyToFloat(lo) - rowmax) * rinv;

        // ---- probs * sigmoid((probs - thr) / SOFTNESS) ----
        float o_[8];
#pragma unroll
        for (int j = 0; j < 8; ++j) {
            float prob = e[j] * rinv;
            float sm   = __frcp_rn(1.0f + __expf(-(prob - thr) * INV_SOFT));
            o_[j] = prob * sm;
        }
        float4* Ov = (float4*)(out + (size_t)row * MDIM + (size_t)tid * 8);
        Ov[0] = make_float4(o_[0], o_[1], o_[2], o_[3]);
        Ov[1] = make_float4(o_[4], o_[5], o_[6], o_[7]);
        // next iteration's top-of-loop barrier separates these LDS reads from
        // the prefetch that will overwrite this buffer.
    }
}

extern "C" void kernel_launch(void* const* d_in, const int* in_sizes, int n_in,
                              void* d_out, int out_size, void* d_ws, size_t ws_size,
                              hipStream_t stream) {
    (void)in_sizes; (void)n_in; (void)out_size; (void)d_ws; (void)ws_size;
    const float* logits = (const float*)d_in[0];
    const float* U      = (const float*)d_in[1];
    float* out          = (float*)d_out;
    dim3 grid(ROWS_ALL / RPB);   // 4096 blocks x 256 threads
    dim3 block(TPB);
    GumbelTopK_kernel<<<grid, block, 0, stream>>>(logits, U, out);
}